// IIA_38491496907265
// MI455X (gfx1250) — compile-verified
//
#include <hip/hip_runtime.h>
#include <math.h>

#define HW 65536
#define WIDTH 256
#define K 480
#define NPROP 30
#define EPS 1.0e-4f

typedef __attribute__((ext_vector_type(2))) float v2f;
typedef __attribute__((ext_vector_type(8))) float v8f;

// ---------------------------------------------------------------------------
// Pass 1: center[p] = clip(sigmoid(dot(features[:,p], w480) + b480))
// GEMV framed as V_WMMA_F32_16X16X4_F32: per wave, 16 pixels, K stepped by 4.
// A (16x4) = weight chunk replicated across all 16 rows; B (4x16) = features.
// All rows of D are identical => acc[0] in lanes 0..15 holds the dot result.
// Bandwidth-bound: 126 MB stream -> ~5.4 us floor @ 23.3 TB/s.
// ---------------------------------------------------------------------------
__global__ __launch_bounds__(256) void center_wmma_kernel(
    const float* __restrict__ feat, const float* __restrict__ weight,
    const float* __restrict__ bias, float* __restrict__ center)
{
    __shared__ float wrow[K];
    for (int i = threadIdx.x; i < K; i += 256)
        wrow[i] = weight[K * K + i];            // row 480 of (481,480)
    __syncthreads();

    const int lane  = threadIdx.x & 31;
    const int wave  = threadIdx.x >> 5;
    const int pix   = (blockIdx.x * 8 + wave) * 16 + (lane & 15);
    const int khalf = (lane < 16) ? 0 : 2;      // ISA 16x4 f32 A layout
    const float* fp = feat + pix;

    v8f acc = {};
    for (int k0 = 0; k0 < K; k0 += 4) {
        const int ka = k0 + khalf;
        v2f a, b;
        a.x = wrow[ka];
        a.y = wrow[ka + 1];
        b.x = fp[(size_t)ka * HW];
        b.y = fp[(size_t)(ka + 1) * HW];
        if (k0 + 4 < K)
            __builtin_prefetch(fp + (size_t)(ka + 4) * HW, 0, 0);
        acc = __builtin_amdgcn_wmma_f32_16x16x4_f32(
            false, a, false, b, (short)0, acc, false, false);
    }

    const float hm = acc[0] + bias[K];
    float s = 1.0f / (1.0f + expf(-hm));
    s = fminf(fmaxf(s, EPS), 1.0f - EPS);
    if (lane < 16) center[pix] = s;
}

// ---------------------------------------------------------------------------
// Pass 2: smoothed = (c + avgpool3x3_zero_pad(c)) * 0.5   (sum/9 always)
// ---------------------------------------------------------------------------
__global__ void smooth_kernel(const float* __restrict__ c, float* __restrict__ s)
{
    const int p = blockIdx.x * 256 + threadIdx.x;
    const int y = p >> 8, x = p & 255;
    float sum = 0.0f;
    for (int dy = -1; dy <= 1; ++dy)
        for (int dx = -1; dx <= 1; ++dx) {
            const int yy = y + dy, xx = x + dx;
            if (yy >= 0 && yy < WIDTH && xx >= 0 && xx < WIDTH)
                sum += c[yy * WIDTH + xx];
        }
    s[p] = (c[p] + sum * (1.0f / 9.0f)) * 0.5f;
}

// ---------------------------------------------------------------------------
// Pass 3: 5x5 max-pool NMS (-inf pad == ignore OOB): keep value iff it is the
// window max, else 0.
// ---------------------------------------------------------------------------
__global__ void nms_kernel(const float* __restrict__ s, float* __restrict__ o)
{
    const int p = blockIdx.x * 256 + threadIdx.x;
    const int y = p >> 8, x = p & 255;
    const float v = s[p];
    float m = -INFINITY;
    for (int dy = -2; dy <= 2; ++dy)
        for (int dx = -2; dx <= 2; ++dx) {
            const int yy = y + dy, xx = x + dx;
            if (yy >= 0 && yy < WIDTH && xx >= 0 && xx < WIDTH)
                m = fmaxf(m, s[yy * WIDTH + xx]);
        }
    o[p] = (m == v) ? v : 0.0f;
}

// ---------------------------------------------------------------------------
// Pass 4: top-30 by 30 sequential argmax passes (tie-break: lowest index,
// matching lax.top_k). Single workgroup; nms map is read-only (L2-resident,
// 256 KB), exclusion list of already-picked indices kept in LDS.
// ---------------------------------------------------------------------------
__global__ __launch_bounds__(1024) void topk_kernel(
    const float* __restrict__ nms, float* __restrict__ topv, int* __restrict__ topi)
{
    __shared__ float sval[1024];
    __shared__ int   sidx[1024];
    __shared__ int   sel[NPROP];
    const int tid = threadIdx.x;

    for (int it = 0; it < NPROP; ++it) {
        float bv = -1.0f;
        int   bi = 0x7fffffff;
        for (int p = tid; p < HW; p += 1024) {
            const float v = nms[p];
            bool taken = false;
            for (int j = 0; j < it; ++j) taken |= (sel[j] == p);
            if (!taken && (v > bv || (v == bv && p < bi))) { bv = v; bi = p; }
        }
        sval[tid] = bv; sidx[tid] = bi;
        __syncthreads();
        for (int off = 512; off > 0; off >>= 1) {
            if (tid < off) {
                const float ov = sval[tid + off];
                const int   oi = sidx[tid + off];
                if (ov > sval[tid] || (ov == sval[tid] && oi < sidx[tid])) {
                    sval[tid] = ov; sidx[tid] = oi;
                }
            }
            __syncthreads();
        }
        if (tid == 0) { topv[it] = sval[0]; topi[it] = sidx[0]; sel[it] = sidx[0]; }
        __syncthreads();
    }
}

// ---------------------------------------------------------------------------
// Pass 5: gather instance params + emit all outputs.
// d_out layout: scores[30] | coords[30*2] (y,x as float) | params[30*480] | valid[30]
// ---------------------------------------------------------------------------
__global__ void gather_kernel(const float* __restrict__ feat,
                              const float* __restrict__ topv,
                              const int* __restrict__ topi,
                              float* __restrict__ out)
{
    const int i = blockIdx.x;            // proposal 0..29
    const int p = topi[i];
    const float sc = topv[i];
    for (int c = threadIdx.x; c < K; c += blockDim.x)
        out[NPROP + 2 * NPROP + i * K + c] = feat[(size_t)c * HW + p];
    if (threadIdx.x == 0) {
        out[i] = sc;
        out[NPROP + 2 * i + 0] = (float)(p >> 8);   // y
        out[NPROP + 2 * i + 1] = (float)(p & 255);  // x
        out[NPROP + 2 * NPROP + NPROP * K + i] = (sc > 0.01f) ? 1.0f : 0.0f;
    }
}

extern "C" void kernel_launch(void* const* d_in, const int* in_sizes, int n_in,
                              void* d_out, int out_size, void* d_ws, size_t ws_size,
                              hipStream_t stream)
{
    const float* feat   = (const float*)d_in[0];   // (1,480,256,256) f32
    const float* weight = (const float*)d_in[1];   // (481,480) f32
    const float* bias   = (const float*)d_in[2];   // (481,) f32
    float* out = (float*)d_out;

    float* ws     = (float*)d_ws;
    float* center = ws;                 // 65536 f32
    float* smooth = ws + HW;            // 65536 f32
    float* nms    = ws + 2 * HW;        // 65536 f32
    float* topv   = ws + 3 * HW;        // 30 f32
    int*   topi   = (int*)(ws + 3 * HW + 32);

    center_wmma_kernel<<<HW / 128, 256, 0, stream>>>(feat, weight, bias, center);
    smooth_kernel<<<HW / 256, 256, 0, stream>>>(center, smooth);
    nms_kernel<<<HW / 256, 256, 0, stream>>>(smooth, nms);
    topk_kernel<<<1, 1024, 0, stream>>>(nms, topv, topi);
    gather_kernel<<<NPROP, 512, 0, stream>>>(feat, topv, topi, out);
}